// HashGridEncoder_2439541424392
// MI455X (gfx1250) — compile-verified
//
#include <hip/hip_runtime.h>
#include <cmath>
#include <cstdint>
#include <cstddef>

// ---------------------------------------------------------------------------
// HashGridEncoder for MI455X (gfx1250)
//  - 2,097,152 points, 16 levels, FEAT=2, trilinear interp, hash tables.
//  - Tables (~47MB) are L2-resident; output written with non-temporal stores
//    so the 256MB output stream does not evict the tables from L2.
//  - Levels 0 and 1 (32KB + 85KB) are staged into LDS with gfx1250 async
//    global->LDS DMA (ASYNCcnt) and gathered with ds_load_b64.
//  - Power-of-two hash tables: 32-bit xor-hash + mask.
//    Non-pow2 (res^3) tables: exact 64-bit hash mod via magic multiplier.
//  - SPEC=true kernel: pow2 pattern {l0, l6..l15} constant-folded, LDS-only
//    for levels 0-1 -> single hash path per level, smaller I$ footprint.
//    Generic fallback kernel used if runtime table sizes mismatch.
// ---------------------------------------------------------------------------

#define NLVL   16
#define PRIME1 2654435761u
#define PRIME2 805459861u

// level-0 / level-1 table entry counts (res 16 -> 4096, res 22 -> 10648)
#define GS0 4096
#define GS1 10648

typedef float f2 __attribute__((ext_vector_type(2)));
typedef float f4 __attribute__((ext_vector_type(4)));
typedef int   v4i __attribute__((vector_size(4 * sizeof(int))));

#if defined(__has_builtin)
#  if __has_builtin(__builtin_amdgcn_global_load_async_to_lds_b128)
#    define HAVE_ASYNC_LDS 1
#  endif
#  if __has_builtin(__builtin_amdgcn_s_wait_asynccnt)
#    define HAVE_WAIT_ASYNC 1
#  endif
#endif

#define AS1 __attribute__((address_space(1)))
#define AS3 __attribute__((address_space(3)))

// Structurally fixed pow2-ness: level0 = 16^3 = 2^12; levels 1..5 = res^3
// (22,30,42,58,80 cubed, none pow2); levels 6..15 hashed at 2^19.
__device__ constexpr bool POW2L[NLVL] = {
    true,  false, false, false, false, false,
    true,  true,  true,  true,  true,  true,  true, true, true, true};

struct LevelP {
  float              resf;   // resolution as float
  int                res;    // resolution
  unsigned           gs;     // table entries
  unsigned           mask;   // gs-1 if power-of-two else 0
  unsigned long long magic;  // floor(2^64/gs)+1  (valid: h < 2^43, gs < 2^21)
};

struct KParams {
  const float* tab[NLVL];
  LevelP       lv[NLVL];
  int          lds_ok;       // 1 if level0/1 table sizes match GS0/GS1
};

__device__ __forceinline__ unsigned hash_mod(unsigned long long h,
                                             unsigned gs,
                                             unsigned long long magic) {
  unsigned long long q = __umul64hi(h, magic);
  return (unsigned)(h - q * (unsigned long long)gs);
}

__device__ __forceinline__ void wait_async0() {
#if defined(HAVE_WAIT_ASYNC)
  __builtin_amdgcn_s_wait_asynccnt((unsigned short)0);
#else
  asm volatile("s_wait_asynccnt 0" ::: "memory");
#endif
}

template <bool SPEC>
__global__ __launch_bounds__(512)
void hashgrid_enc_kernel(KParams P, const float* __restrict__ xyz,
                         float* __restrict__ out, int npts) {
  __shared__ f4 lds0r[GS0 / 2];   // 4096 float2 = 32 KB
  __shared__ f4 lds1r[GS1 / 2];   // 10648 float2 = 85.2 KB

  // ---- stage levels 0,1 into LDS (async DMA path on gfx1250) ----
  if (SPEC || P.lds_ok) {
    const f4* s0 = (const f4*)P.tab[0];
    const f4* s1 = (const f4*)P.tab[1];
#if defined(HAVE_ASYNC_LDS)
    for (int i = threadIdx.x; i < GS0 / 2; i += 512)
      __builtin_amdgcn_global_load_async_to_lds_b128(
          (AS1 v4i*)(s0 + i), (AS3 v4i*)(lds0r + i), 0, 0);
    for (int i = threadIdx.x; i < GS1 / 2; i += 512)
      __builtin_amdgcn_global_load_async_to_lds_b128(
          (AS1 v4i*)(s1 + i), (AS3 v4i*)(lds1r + i), 0, 0);
    wait_async0();
#else
    for (int i = threadIdx.x; i < GS0 / 2; i += 512) lds0r[i] = s0[i];
    for (int i = threadIdx.x; i < GS1 / 2; i += 512) lds1r[i] = s1[i];
#endif
  }
  __syncthreads();

  const f2* lt0 = (const f2*)lds0r;
  const f2* lt1 = (const f2*)lds1r;

  const int stride = gridDim.x * blockDim.x;
  for (int p = blockIdx.x * blockDim.x + threadIdx.x; p < npts; p += stride) {
    const float* xp = xyz + (size_t)p * 3;
    float x = __builtin_nontemporal_load(xp + 0);
    float y = __builtin_nontemporal_load(xp + 1);
    float z = __builtin_nontemporal_load(xp + 2);
    // xn = clip((xyz + 1) / 2, 0, 1)
    float xn0 = fminf(fmaxf((x + 1.0f) * 0.5f, 0.0f), 1.0f);
    float xn1 = fminf(fmaxf((y + 1.0f) * 0.5f, 0.0f), 1.0f);
    float xn2 = fminf(fmaxf((z + 1.0f) * 0.5f, 0.0f), 1.0f);

    float acc[2 * NLVL];

#pragma unroll
    for (int l = 0; l < NLVL; ++l) {
      const LevelP lp = P.lv[l];
      float gx = xn0 * lp.resf;
      float gy = xn1 * lp.resf;
      float gz = xn2 * lp.resf;
      float fxf = floorf(gx), fyf = floorf(gy), fzf = floorf(gz);
      float fx = gx - fxf, fy = gy - fyf, fz = gz - fzf;
      int ix = (int)fxf, iy = (int)fyf, iz = (int)fzf;
      int rm1 = lp.res - 1;
      // g >= 0 so floor >= 0; only upper clamps needed
      unsigned x0 = (unsigned)min(ix, rm1),     y0 = (unsigned)min(iy, rm1),
               z0 = (unsigned)min(iz, rm1);
      unsigned x1 = (unsigned)min(ix + 1, lp.res), y1 = (unsigned)min(iy + 1, lp.res),
               z1 = (unsigned)min(iz + 1, lp.res);

      unsigned idx[8];
      const bool use_pow2 = SPEC ? POW2L[l] : (lp.mask != 0u);
      if (use_pow2) {
        // power-of-two table: only low bits of the int64 hash survive the mask
        unsigned ya = y0 * PRIME1, yb = y1 * PRIME1;
        unsigned za = z0 * PRIME2, zb = z1 * PRIME2;
        idx[0] = (x0 ^ ya ^ za) & lp.mask;
        idx[1] = (x1 ^ ya ^ za) & lp.mask;
        idx[2] = (x0 ^ yb ^ za) & lp.mask;
        idx[3] = (x1 ^ yb ^ za) & lp.mask;
        idx[4] = (x0 ^ ya ^ zb) & lp.mask;
        idx[5] = (x1 ^ ya ^ zb) & lp.mask;
        idx[6] = (x0 ^ yb ^ zb) & lp.mask;
        idx[7] = (x1 ^ yb ^ zb) & lp.mask;
      } else {
        // exact 64-bit hash, modulo via magic multiplier (h < 2^43, gs < 2^21)
        unsigned long long ya = (unsigned long long)y0 * PRIME1;
        unsigned long long yb = (unsigned long long)y1 * PRIME1;
        unsigned long long za = (unsigned long long)z0 * PRIME2;
        unsigned long long zb = (unsigned long long)z1 * PRIME2;
        unsigned long long xa = x0, xb = x1;
        idx[0] = hash_mod(xa ^ ya ^ za, lp.gs, lp.magic);
        idx[1] = hash_mod(xb ^ ya ^ za, lp.gs, lp.magic);
        idx[2] = hash_mod(xa ^ yb ^ za, lp.gs, lp.magic);
        idx[3] = hash_mod(xb ^ yb ^ za, lp.gs, lp.magic);
        idx[4] = hash_mod(xa ^ ya ^ zb, lp.gs, lp.magic);
        idx[5] = hash_mod(xb ^ ya ^ zb, lp.gs, lp.magic);
        idx[6] = hash_mod(xa ^ yb ^ zb, lp.gs, lp.magic);
        idx[7] = hash_mod(xb ^ yb ^ zb, lp.gs, lp.magic);
      }

      f2 cv[8];
      if (l < 2 && (SPEC || P.lds_ok)) {
        const f2* lt = (l == 0) ? lt0 : lt1;   // LDS gathers (ds_load_b64)
#pragma unroll
        for (int c = 0; c < 8; ++c) cv[c] = lt[idx[c]];
      } else {
        const f2* t = (const f2*)P.tab[l];     // global gathers (L2-resident)
#pragma unroll
        for (int c = 0; c < 8; ++c) cv[c] = t[idx[c]];
      }

      float wx0 = 1.0f - fx, wx1 = fx;
      float wy0 = 1.0f - fy, wy1 = fy;
      float wz0 = 1.0f - fz, wz1 = fz;
      float w00 = wy0 * wz0, w10 = wy1 * wz0;
      float w01 = wy0 * wz1, w11 = wy1 * wz1;

      float f0 = 0.0f, f1 = 0.0f, w;
      w = wx0 * w00; f0 = fmaf(w, cv[0].x, f0); f1 = fmaf(w, cv[0].y, f1);
      w = wx1 * w00; f0 = fmaf(w, cv[1].x, f0); f1 = fmaf(w, cv[1].y, f1);
      w = wx0 * w10; f0 = fmaf(w, cv[2].x, f0); f1 = fmaf(w, cv[2].y, f1);
      w = wx1 * w10; f0 = fmaf(w, cv[3].x, f0); f1 = fmaf(w, cv[3].y, f1);
      w = wx0 * w01; f0 = fmaf(w, cv[4].x, f0); f1 = fmaf(w, cv[4].y, f1);
      w = wx1 * w01; f0 = fmaf(w, cv[5].x, f0); f1 = fmaf(w, cv[5].y, f1);
      w = wx0 * w11; f0 = fmaf(w, cv[6].x, f0); f1 = fmaf(w, cv[6].y, f1);
      w = wx1 * w11; f0 = fmaf(w, cv[7].x, f0); f1 = fmaf(w, cv[7].y, f1);

      acc[2 * l + 0] = f0;
      acc[2 * l + 1] = f1;
    }

    // 32 contiguous floats per point; non-temporal so L2 keeps the tables
    f4* op = (f4*)(out + (size_t)p * 32);
#pragma unroll
    for (int i = 0; i < 8; ++i) {
      f4 v = {acc[4 * i + 0], acc[4 * i + 1], acc[4 * i + 2], acc[4 * i + 3]};
      __builtin_nontemporal_store(v, op + i);
    }
  }
}

extern "C" void kernel_launch(void* const* d_in, const int* in_sizes, int n_in,
                              void* d_out, int out_size, void* d_ws, size_t ws_size,
                              hipStream_t stream) {
  (void)d_ws; (void)ws_size; (void)out_size; (void)n_in;

  const float* xyz = (const float*)d_in[0];
  const int npts = in_sizes[0] / 3;

  // host mirror of the device POW2L pattern
  const bool pow2_expect[NLVL] = {true,  false, false, false, false, false,
                                  true,  true,  true,  true,  true,  true,
                                  true,  true,  true,  true};

  KParams P;
  // Replicate numpy-float32 resolution math:
  //   b = exp2f(log2f(128.f)/15.f); res_l = (int)(16.f * powf(b, l))
  const float  cexp = 7.0f / 15.0f;                 // log2f(128)/15 in fp32
  const double bgrow = exp2((double)cexp);          // correctly-rounded fp32 path
  const float  bf = (float)bgrow;

  bool spec_ok = true;
  for (int l = 0; l < NLVL; ++l) {
    unsigned gs = (unsigned)(in_sizes[1 + l] / 2);  // entries = elems / FEAT
    float pw  = (float)pow((double)bf, (double)l);  // matches fp32 powf
    int   res = (int)(16.0f * pw);
    if (gs != 524288u) {
      // small level: gs == res^3 exactly -> recover res from ground truth
      int r = (int)llround(cbrt((double)gs));
      while ((long long)r * r * r > (long long)gs) --r;
      while ((long long)(r + 1) * (r + 1) * (r + 1) <= (long long)gs) ++r;
      if ((long long)r * r * r == (long long)gs) res = r;
    }
    P.tab[l]      = (const float*)d_in[1 + l];
    P.lv[l].resf  = (float)res;
    P.lv[l].res   = res;
    P.lv[l].gs    = gs;
    P.lv[l].mask  = ((gs & (gs - 1)) == 0u) ? (gs - 1u) : 0u;
    P.lv[l].magic = (~0ULL / (unsigned long long)gs) + 1ULL;
    if ((P.lv[l].mask != 0u) != pow2_expect[l]) spec_ok = false;
  }
  P.lds_ok = (P.lv[0].gs == (unsigned)GS0 && P.lv[1].gs == (unsigned)GS1) ? 1 : 0;
  if (!P.lds_ok) spec_ok = false;

  dim3 grid(1024), block(512);
  if (spec_ok) {
    hipLaunchKernelGGL((hashgrid_enc_kernel<true>), grid, block, 0, stream,
                       P, xyz, (float*)d_out, npts);
  } else {
    hipLaunchKernelGGL((hashgrid_enc_kernel<false>), grid, block, 0, stream,
                       P, xyz, (float*)d_out, npts);
  }
}